// CommitGCN_28750511080087
// MI455X (gfx1250) — compile-verified
//
#include <hip/hip_runtime.h>
#include <hip/hip_bf16.h>

typedef __attribute__((ext_vector_type(2))) float v2f;
typedef __attribute__((ext_vector_type(8))) float v8f;

#define F_IN 128
#define HID 64

// ---------------- degree / normalization ----------------
__global__ void deg_count_kernel(const int* __restrict__ col, float* __restrict__ deg, long E) {
    long t = (long)blockIdx.x * blockDim.x + threadIdx.x;
    if (t < E) atomicAdd(&deg[col[t]], 1.0f);
}

__global__ void deg_to_dis_kernel(float* __restrict__ deg, int n) {
    int i = blockIdx.x * blockDim.x + threadIdx.x;
    if (i < n) deg[i] = rsqrtf(deg[i] + 1.0f);  // +1 = self loop; deg>=1 always
}

// ---------------- fp32 WMMA GEMM: H[M x 64] = A[M x KDIM] @ W[KDIM x 64] ----------------
// One wave computes a 16-row tile across all 64 output columns.
// A-frag (16x4 f32): v0 = {lanes0-15: K=k0, lanes16-31: K=k0+2}, v1 = {K=k0+1, K=k0+3}
// B-frag (4x16 f32): same K placement, N = lane&15 striped across lanes.
// W is staged in LDS pre-paired along K: sW2[kp*64+n] = {W[2kp][n], W[2kp+1][n]}
// so each B-fragment is ONE aligned ds_load_b64 straight into an even VGPR pair.
template <int KDIM>
__global__ void gemm_wmma_kernel(const float* __restrict__ A, const float* __restrict__ W,
                                 float* __restrict__ H, int M) {
    __shared__ float2 sW2[(KDIM / 2) * HID];
    for (int i = threadIdx.y * 32 + threadIdx.x; i < (KDIM / 2) * HID; i += 32 * blockDim.y) {
        int kp = i >> 6, n = i & 63;
        sW2[i] = make_float2(W[(2 * kp) * HID + n], W[(2 * kp + 1) * HID + n]);
    }
    __syncthreads();

    int mTile = blockIdx.x * blockDim.y + threadIdx.y;
    int mBase = mTile * 16;
    if (mBase >= M) return;  // uniform per wave -> EXEC stays all-ones for WMMA

    const int lane  = threadIdx.x;
    const int ncol  = lane & 15;
    const int kHalf = (lane >> 4);          // 0 for lanes 0-15, 1 for lanes 16-31
    const int kOff  = kHalf * 2;            // K offset within the 4-wide step
    const float* aPtr = A + (long)(mBase + ncol) * KDIM + kOff;
    const float2* bPtr = sW2 + kHalf * HID + ncol;

    v8f acc0 = {}, acc1 = {}, acc2 = {}, acc3 = {};
#pragma unroll 4
    for (int k0 = 0; k0 < KDIM; k0 += 4) {
        const float2 av = *(const float2*)(aPtr + k0);          // global_load_b64
        v2f a; a.x = av.x; a.y = av.y;
        const float2* bRow = bPtr + (k0 >> 1) * HID;
        float2 b0 = bRow[0];                                    // ds_load_b64 each
        float2 b1 = bRow[16];
        float2 b2 = bRow[32];
        float2 b3 = bRow[48];
        v2f b;
        b.x = b0.x; b.y = b0.y;
        acc0 = __builtin_amdgcn_wmma_f32_16x16x4_f32(false, a, false, b, (short)0, acc0, false, false);
        b.x = b1.x; b.y = b1.y;
        acc1 = __builtin_amdgcn_wmma_f32_16x16x4_f32(false, a, false, b, (short)0, acc1, false, false);
        b.x = b2.x; b.y = b2.y;
        acc2 = __builtin_amdgcn_wmma_f32_16x16x4_f32(false, a, false, b, (short)0, acc2, false, false);
        b.x = b3.x; b.y = b3.y;
        acc3 = __builtin_amdgcn_wmma_f32_16x16x4_f32(false, a, false, b, (short)0, acc3, false, false);
    }

    // C/D layout: VGPR v, lanes0-15 -> row mBase+v, lanes16-31 -> row mBase+8+v, col = lane&15
    const int rowOff = kHalf * 8;
#pragma unroll
    for (int v = 0; v < 8; ++v) {
        long r = (long)(mBase + rowOff + v) * HID + ncol;
        H[r +  0] = acc0[v];
        H[r + 16] = acc1[v];
        H[r + 32] = acc2[v];
        H[r + 48] = acc3[v];
    }
}

// ---------------- edge scatter: agg[col] += h[row] * dis[row]*dis[col] ----------------
__global__ void scatter_kernel(const float* __restrict__ h, const int* __restrict__ row,
                               const int* __restrict__ col, const float* __restrict__ dis,
                               float* __restrict__ agg, long E) {
    long t = (long)blockIdx.x * blockDim.x + threadIdx.x;
    long e = t >> 5;
    if (e >= E) return;
    int f = (int)(t & 31) * 2;           // 2 features per lane, 32 lanes = 64 feats/edge
    int r = row[e], c = col[e];
    float norm = dis[r] * dis[c];        // same addr across wave -> single L2 hit
    const float2 hv = *(const float2*)(h + (long)r * HID + f);
    float* dst = agg + (long)c * HID + f;
    atomicAdd(dst,     hv.x * norm);
    atomicAdd(dst + 1, hv.y * norm);
}

// ---------------- finalize: agg = relu(agg + h*dis^2 + b) (self-loop + bias + relu) ----------------
__global__ void finalize_kernel(float* __restrict__ agg, const float* __restrict__ h,
                                const float* __restrict__ dis, const float* __restrict__ bias,
                                int n) {
    long t = (long)blockIdx.x * blockDim.x + threadIdx.x;
    if (t >= (long)n * HID) return;
    int i = (int)(t >> 6), f = (int)(t & 63);
    float d = dis[i];
    float v = agg[t] + h[t] * d * d + bias[f];
    agg[t] = v > 0.0f ? v : 0.0f;
}

// ---------------- pooling + FC ----------------
__global__ void pool_kernel(const float* __restrict__ h, const int* __restrict__ batch,
                            float* __restrict__ sums, float* __restrict__ counts, int n) {
    long t = (long)blockIdx.x * blockDim.x + threadIdx.x;
    if (t >= (long)n * HID) return;
    int i = (int)(t >> 6), f = (int)(t & 63);
    int g = batch[i];
    atomicAdd(&sums[(long)g * HID + f], h[t]);
    if (f == 0) atomicAdd(&counts[g], 1.0f);
}

__global__ void fc_kernel(const float* __restrict__ sums, const float* __restrict__ counts,
                          const float* __restrict__ Wfc, const float* __restrict__ bfc,
                          float* __restrict__ out, int num_graphs, int num_classes) {
    int t = blockIdx.x * blockDim.x + threadIdx.x;
    if (t >= num_graphs * num_classes) return;
    int g = t / num_classes, c = t % num_classes;
    float inv = 1.0f / fmaxf(counts[g], 1.0f);
    float acc = 0.0f;
#pragma unroll
    for (int k = 0; k < HID; ++k)
        acc += sums[(long)g * HID + k] * Wfc[k * num_classes + c];
    out[t] = acc * inv + bfc[c];
}

extern "C" void kernel_launch(void* const* d_in, const int* in_sizes, int n_in,
                              void* d_out, int out_size, void* d_ws, size_t ws_size,
                              hipStream_t stream) {
    const float* x    = (const float*)d_in[0];
    const int*   ei   = (const int*)d_in[1];   // edge_index flattened (2,E): row then col
    const int*   batch= (const int*)d_in[2];
    const float* W1   = (const float*)d_in[3];
    const float* b1   = (const float*)d_in[4];
    const float* W2   = (const float*)d_in[5];
    const float* b2   = (const float*)d_in[6];
    const float* Wfc  = (const float*)d_in[7];
    const float* bfc  = (const float*)d_in[8];
    float* out = (float*)d_out;

    const int  n = in_sizes[0] / F_IN;       // 100000
    const long E = in_sizes[1] / 2;          // 3200000
    const int  num_classes = in_sizes[8];    // 2
    const int  num_graphs  = out_size / num_classes;  // 512
    const int* rowI = ei;
    const int* colI = ei + E;

    // workspace layout (floats)
    float* bufA   = (float*)d_ws;                 // [n*HID]  x@W products
    float* bufB   = bufA + (long)n * HID;         // [n*HID]  aggregation / layer outputs
    float* dis    = bufB + (long)n * HID;         // [n]      deg -> dis (in place)
    float* sums   = dis + n;                      // [G*HID]
    float* counts = sums + (long)num_graphs * HID;// [G]
    size_t need = ((size_t)(counts + num_graphs) - (size_t)d_ws);
    if (ws_size < need) return;

    const int TPB = 256;
    const int mTiles = (n + 15) / 16;
    const int wavesPerBlk = 8;
    dim3 gemmBlk(32, wavesPerBlk);
    int  gemmGrid = (mTiles + wavesPerBlk - 1) / wavesPerBlk;
    long nh = (long)n * HID;

    // ---- degree / dis ----
    hipMemsetAsync(dis, 0, (size_t)n * sizeof(float), stream);
    deg_count_kernel<<<(int)((E + TPB - 1) / TPB), TPB, 0, stream>>>(colI, dis, E);
    deg_to_dis_kernel<<<(n + TPB - 1) / TPB, TPB, 0, stream>>>(dis, n);

    // ---- layer 1: h1 = x@W1 ; agg ; relu(agg + self + b1) -> bufB ----
    gemm_wmma_kernel<F_IN><<<gemmGrid, gemmBlk, 0, stream>>>(x, W1, bufA, n);
    hipMemsetAsync(bufB, 0, (size_t)nh * sizeof(float), stream);
    scatter_kernel<<<(int)((E * 32 + TPB - 1) / TPB), TPB, 0, stream>>>(bufA, rowI, colI, dis, bufB, E);
    finalize_kernel<<<(int)((nh + TPB - 1) / TPB), TPB, 0, stream>>>(bufB, bufA, dis, b1, n);

    // ---- layer 2: h2 = out1@W2 -> bufA ; agg -> bufB ; relu -> bufB ----
    gemm_wmma_kernel<HID><<<gemmGrid, gemmBlk, 0, stream>>>(bufB, W2, bufA, n);
    hipMemsetAsync(bufB, 0, (size_t)nh * sizeof(float), stream);
    scatter_kernel<<<(int)((E * 32 + TPB - 1) / TPB), TPB, 0, stream>>>(bufA, rowI, colI, dis, bufB, E);
    finalize_kernel<<<(int)((nh + TPB - 1) / TPB), TPB, 0, stream>>>(bufB, bufA, dis, b2, n);

    // ---- pooling + fc ----
    hipMemsetAsync(sums, 0, (size_t)(num_graphs * HID + num_graphs) * sizeof(float), stream);
    pool_kernel<<<(int)((nh + TPB - 1) / TPB), TPB, 0, stream>>>(bufB, batch, sums, counts, n);
    fc_kernel<<<(num_graphs * num_classes + TPB - 1) / TPB, TPB, 0, stream>>>(
        sums, counts, Wfc, bfc, out, num_graphs, num_classes);
}